// ChamferDistanceLoss_56882546868324
// MI455X (gfx1250) — compile-verified
//
#include <hip/hip_runtime.h>

typedef float v2f __attribute__((ext_vector_type(2)));
typedef float v4f __attribute__((ext_vector_type(4)));
typedef float v8f __attribute__((ext_vector_type(8)));

#define N_PTS   8192
#define TILE    16
#define WPB     4                       // waves per block
#define UNROLL  8
#define COL_TILES (N_PTS / TILE)        // 512
#define ROW_TILES (N_PTS / TILE)        // 512
#define BLOCKS_X  (ROW_TILES / WPB)     // 128
#define N_PARTIAL (ROW_TILES * 4 * 2)   // 4096 per-wave partials
#define PTS_PER_SET (4 * N_PTS)         // 32768 points per array

__device__ __forceinline__ float min3f(float a, float b, float c) {
    float r;
    asm("v_min3_num_f32 %0, %1, %2, %3" : "=v"(r) : "v"(a), "v"(b), "v"(c));
    return r;
}

// Precompute embedded column fragments: e4[p] = {-2x, -2y, -2z, |p|^2}
__global__ __launch_bounds__(256)
void embed_kernel(const float* __restrict__ tpl, const float* __restrict__ src,
                  float* __restrict__ e4T, float* __restrict__ e4S)
{
    const int i = blockIdx.x * 256 + threadIdx.x;
    if (i >= PTS_PER_SET) return;
    {
        const float x = tpl[3 * i + 0], y = tpl[3 * i + 1], z = tpl[3 * i + 2];
        v4f e; e.x = -2.0f * x; e.y = -2.0f * y; e.z = -2.0f * z;
        e.w = x * x + y * y + z * z;
        *(v4f*)(e4T + 4 * (size_t)i) = e;
    }
    {
        const float x = src[3 * i + 0], y = src[3 * i + 1], z = src[3 * i + 2];
        v4f e; e.x = -2.0f * x; e.y = -2.0f * y; e.z = -2.0f * z;
        e.w = x * x + y * y + z * z;
        *(v4f*)(e4S + 4 * (size_t)i) = e;
    }
}

__global__ __launch_bounds__(WPB * 32)
void chamfer_dir_kernel(const float* __restrict__ tpl,
                        const float* __restrict__ src,
                        const float* __restrict__ e4T,
                        const float* __restrict__ e4S,
                        float* __restrict__ partial)
{
    const int lane  = threadIdx.x & 31;
    const int wave  = threadIdx.x >> 5;
    const int batch = blockIdx.y;
    const int dir   = blockIdx.z;

    // dir 0: rows = template, cols = source (embedded).  dir 1: swapped.
    const float* p0  = (dir == 0) ? tpl : src;    // raw rows
    const float* e4c = (dir == 0) ? e4S : e4T;    // embedded cols
    p0  += (size_t)batch * N_PTS * 3;
    e4c += (size_t)batch * N_PTS * 4;

    const int rowTile = blockIdx.x * WPB + wave;   // 0..511
    const int c16     = lane & 15;
    const bool hi     = lane >= 16;

    // ---- A fragment (persistent): row point embedded as (tx,ty,tz,1) ----
    // 32-bit A 16x4 layout: lanes 0-15 -> K=0,1 ; lanes 16-31 -> K=2,3.
    const int r = rowTile * TILE + c16;
    const float tx = p0[3 * r + 0];
    const float ty = p0[3 * r + 1];
    const float tz = p0[3 * r + 2];
    const float tsq = tx * tx + ty * ty + tz * tz;
    v2f a;
    a.x = hi ? tz   : tx;
    a.y = hi ? 1.0f : ty;

    // Per-lane column base: lane<16 reads dwords {0,1} (K0,K1) of its column
    // point, lane>=16 reads dwords {2,3} (K2,K3). Tile stride = 64 floats.
    const float* cb = e4c + (size_t)c16 * 4 + (hi ? 2 : 0);

    v8f m;
    #pragma unroll
    for (int k = 0; k < 8; ++k) m[k] = 1e30f;

    const v8f czero = {};

    for (int j = 0; j < COL_TILES; j += UNROLL) {
        v2f bf[UNROLL];
        #pragma unroll
        for (int u = 0; u < UNROLL; ++u)
            bf[u] = *(const v2f*)(cb + (size_t)(j + u) * 64);

        v8f d[UNROLL];
        #pragma unroll
        for (int u = 0; u < UNROLL; ++u)
            d[u] = __builtin_amdgcn_wmma_f32_16x16x4_f32(
                false, a, false, bf[u], (short)0, czero, false, false);

        #pragma unroll
        for (int k = 0; k < 8; ++k) {
            const float t0 = min3f(d[0][k], d[1][k], d[2][k]);
            const float t1 = min3f(d[3][k], d[4][k], d[5][k]);
            const float t2 = min3f(d[6][k], d[7][k], t0);
            m[k] = min3f(t1, t2, m[k]);
        }
    }

    // ---- finalize: d_min(row) = |t_row|^2 + min_cols(m) ; sum the 16 rows ----
    // C/D layout: element (row = v + 8*hi, col = lane%16) in VGPR v.
    float s = 0.0f;
    #pragma unroll
    for (int v = 0; v < 8; ++v) {
        const int row = v + (hi ? 8 : 0);
        const float t2 = __shfl(tsq, row, 32);         // row norms live on lanes 0-15
        float x = m[v] + t2;
        x = fminf(x, __shfl_xor(x, 8, 32));            // min across the 16-lane half
        x = fminf(x, __shfl_xor(x, 4, 32));
        x = fminf(x, __shfl_xor(x, 2, 32));
        x = fminf(x, __shfl_xor(x, 1, 32));
        s += x;                                        // rows v (half0) / v+8 (half1)
    }
    s += __shfl_xor(s, 16, 32);                        // combine halves: 16 rows total

    if (lane == 0) {
        const int idx = (dir * 4 + batch) * ROW_TILES + rowTile;
        partial[idx] = s;
    }
}

__global__ __launch_bounds__(256)
void chamfer_finalize(const float* __restrict__ partial, float* __restrict__ out)
{
    __shared__ float sm[256];
    float s = 0.0f;
    for (int i = threadIdx.x; i < N_PARTIAL; i += 256) s += partial[i];
    sm[threadIdx.x] = s;
    __syncthreads();
    for (int off = 128; off > 0; off >>= 1) {
        if (threadIdx.x < off) sm[threadIdx.x] += sm[threadIdx.x + off];
        __syncthreads();
    }
    if (threadIdx.x == 0)
        out[0] = sm[0] * (1.0f / (float)(N_PTS * 4));  // mean over N then over batches
}

extern "C" void kernel_launch(void* const* d_in, const int* in_sizes, int n_in,
                              void* d_out, int out_size, void* d_ws, size_t ws_size,
                              hipStream_t stream)
{
    const float* tpl = (const float*)d_in[0];   // template (4,8192,3)
    const float* src = (const float*)d_in[1];   // source   (4,8192,3)
    float* out = (float*)d_out;                 // scalar

    // Workspace layout: [partials: 4096 f][e4T: 512KB][e4S: 512KB]
    float* partial = (float*)d_ws;
    float* e4T = (float*)((char*)d_ws + 16 * 1024);
    float* e4S = e4T + (size_t)PTS_PER_SET * 4;

    embed_kernel<<<(PTS_PER_SET + 255) / 256, 256, 0, stream>>>(tpl, src, e4T, e4S);

    dim3 grid(BLOCKS_X, 4, 2);
    chamfer_dir_kernel<<<grid, WPB * 32, 0, stream>>>(tpl, src, e4T, e4S, partial);

    chamfer_finalize<<<1, 256, 0, stream>>>(partial, out);
}